// TWPGraphConv_37056977830254
// MI455X (gfx1250) — compile-verified
//
#include <hip/hip_runtime.h>
#include <math.h>

typedef float v2f __attribute__((ext_vector_type(2)));
typedef float v8f __attribute__((ext_vector_type(8)));

#define IN_FEATS 128
#define OUT_FEATS 256
#define NEG_SLOPE 0.2f

// register blocking: each wave computes (MB*16) x (NB*16) of the output
#define MB 5   // 3125 M-tiles = 5^5 -> divisible by 5, no tail
#define NB 2

// ---------------------------------------------------------------------------
// Blocked WMMA fp32 GEMM core.
// A row-major [n, K] (K = 128); Wt is W transposed: Wt[col*K + k] = W[k*OUT + col]
// V_WMMA_F32_16X16X4_F32 lane layout:
//   A frag : lane L<16 -> row M=L, K=kk,kk+1 ; lane L+16 -> same row, K=kk+2,kk+3
//   B frag : lane L<16 -> col N=L, K=kk,kk+1 ; lane L+16 -> same col, K=kk+2,kk+3
//   C/D    : vgpr r, lane L<16 -> C[r][L] ; lane L>=16 -> C[r+8][L-16]
// Both A and B fragments are contiguous v2f (8B aligned) loads thanks to Wt.
// ---------------------------------------------------------------------------
__device__ __forceinline__ void wmma_block_f32(const float* __restrict__ A,
                                               const float* __restrict__ Wt,
                                               int rowBase, int colBase,
                                               v8f acc[MB][NB]) {
    const int lane = threadIdx.x & 31;
    const int half = lane >> 4;   // 0 or 1
    const int l16  = lane & 15;

    const float* arow[MB];
#pragma unroll
    for (int m = 0; m < MB; ++m)
        arow[m] = A + (size_t)(rowBase + 16 * m + l16) * IN_FEATS;
    const float* brow[NB];
#pragma unroll
    for (int j = 0; j < NB; ++j)
        brow[j] = Wt + (size_t)(colBase + 16 * j + l16) * IN_FEATS;

    for (int k0 = 0; k0 < IN_FEATS; k0 += 4) {
        const int kk = k0 + 2 * half;
        v2f a[MB], b[NB];
#pragma unroll
        for (int m = 0; m < MB; ++m)
            a[m] = *(const v2f*)(arow[m] + kk);
#pragma unroll
        for (int j = 0; j < NB; ++j)
            b[j] = *(const v2f*)(brow[j] + kk);
#pragma unroll
        for (int m = 0; m < MB; ++m)
#pragma unroll
            for (int j = 0; j < NB; ++j)
                acc[m][j] = __builtin_amdgcn_wmma_f32_16x16x4_f32(
                    false, a[m], false, b[j], (short)0, acc[m][j], false, false);
    }
}

// ---------------------------------------------------------------------------
// init workspace: zero degrees/sums/h, seg_max = -inf
// ---------------------------------------------------------------------------
__global__ void init_ws_kernel(float* norm_l, float* norm_r, float* seg_max,
                               float* seg_sum, float* h, int n, int hsize) {
    int i = blockIdx.x * blockDim.x + threadIdx.x;
    if (i < hsize) h[i] = 0.f;
    if (i < n) {
        norm_l[i] = 0.f;   // accumulates out_deg first
        norm_r[i] = 0.f;   // accumulates in_deg first
        seg_sum[i] = 0.f;
        seg_max[i] = -INFINITY;
    }
}

// W [128,256] row-major -> Wt [256,128] row-major (Wt[c*128+k] = W[k*256+c])
__global__ void transpose_w_kernel(const float* __restrict__ W, float* __restrict__ Wt) {
    int idx = blockIdx.x * blockDim.x + threadIdx.x;   // 0 .. 128*256-1
    if (idx >= IN_FEATS * OUT_FEATS) return;
    int k = idx / OUT_FEATS;
    int c = idx % OUT_FEATS;
    Wt[(size_t)c * IN_FEATS + k] = W[idx];
}

__global__ void degree_kernel(const int* __restrict__ src, const int* __restrict__ dst,
                              float* out_deg, float* in_deg, int E) {
    int e = blockIdx.x * blockDim.x + threadIdx.x;
    if (e >= E) return;
    atomicAdd(out_deg + src[e], 1.f);
    atomicAdd(in_deg + dst[e], 1.f);
}

__global__ void norm_kernel(float* norm_l, float* norm_r, int n) {
    int i = blockIdx.x * blockDim.x + threadIdx.x;
    if (i >= n) return;
    float od = norm_l[i]; od = od < 1.f ? 1.f : od;
    float id = norm_r[i]; id = id < 1.f ? 1.f : id;
    norm_l[i] = rsqrtf(od);
    norm_r[i] = rsqrtf(id);
}

// ---------------------------------------------------------------------------
// projection GEMM: P = feat @ W ; w_src = norm_l[row] * P ; w_dst = tanh(P)
// one wave per 80x32 output block
// ---------------------------------------------------------------------------
__global__ void gemm_proj_kernel(const float* __restrict__ feat, const float* __restrict__ Wt,
                                 const float* __restrict__ norm_l,
                                 float* __restrict__ w_src, float* __restrict__ w_dst,
                                 int n) {
    const int wave    = (blockIdx.x * blockDim.x + threadIdx.x) >> 5;
    const int nBlocks = OUT_FEATS / (16 * NB);                 // 8
    const int mb = wave / nBlocks;
    const int nb = wave % nBlocks;
    const int rowBase = mb * 16 * MB;
    const int colBase = nb * 16 * NB;
    if (rowBase >= n) return;

    v8f acc[MB][NB];
#pragma unroll
    for (int m = 0; m < MB; ++m)
#pragma unroll
        for (int j = 0; j < NB; ++j)
            acc[m][j] = (v8f){0.f, 0.f, 0.f, 0.f, 0.f, 0.f, 0.f, 0.f};

    wmma_block_f32(feat, Wt, rowBase, colBase, acc);

    const int lane = threadIdx.x & 31;
    const int half = lane >> 4;
    const int l16  = lane & 15;
#pragma unroll
    for (int m = 0; m < MB; ++m) {
#pragma unroll
        for (int j = 0; j < NB; ++j) {
            const int col = colBase + 16 * j + l16;
#pragma unroll
            for (int r = 0; r < 8; ++r) {
                const int row = rowBase + 16 * m + (half ? r + 8 : r);
                const float p = acc[m][j][r];
                w_dst[(size_t)row * OUT_FEATS + col] = tanhf(p);
                w_src[(size_t)row * OUT_FEATS + col] = p * norm_l[row];
            }
        }
    }
}

// ---------------------------------------------------------------------------
// output GEMM: rst = (h @ W) * norm_r[row]
// ---------------------------------------------------------------------------
__global__ void gemm_out_kernel(const float* __restrict__ h, const float* __restrict__ Wt,
                                const float* __restrict__ norm_r,
                                float* __restrict__ rst, int n) {
    const int wave    = (blockIdx.x * blockDim.x + threadIdx.x) >> 5;
    const int nBlocks = OUT_FEATS / (16 * NB);
    const int mb = wave / nBlocks;
    const int nb = wave % nBlocks;
    const int rowBase = mb * 16 * MB;
    const int colBase = nb * 16 * NB;
    if (rowBase >= n) return;

    v8f acc[MB][NB];
#pragma unroll
    for (int m = 0; m < MB; ++m)
#pragma unroll
        for (int j = 0; j < NB; ++j)
            acc[m][j] = (v8f){0.f, 0.f, 0.f, 0.f, 0.f, 0.f, 0.f, 0.f};

    wmma_block_f32(h, Wt, rowBase, colBase, acc);

    const int lane = threadIdx.x & 31;
    const int half = lane >> 4;
    const int l16  = lane & 15;
#pragma unroll
    for (int m = 0; m < MB; ++m) {
#pragma unroll
        for (int j = 0; j < NB; ++j) {
            const int col = colBase + 16 * j + l16;
#pragma unroll
            for (int r = 0; r < 8; ++r) {
                const int row = rowBase + 16 * m + (half ? r + 8 : r);
                rst[(size_t)row * OUT_FEATS + col] = acc[m][j][r] * norm_r[row];
            }
        }
    }
}

// ---------------------------------------------------------------------------
// edge attention scores: one wave per edge; 256-wide dot + leaky relu + atomic max
// ---------------------------------------------------------------------------
__device__ __forceinline__ void atomicMaxFloat(float* addr, float v) {
    if (v >= 0.f) atomicMax((int*)addr, __float_as_int(v));
    else          atomicMin((unsigned int*)addr, __float_as_uint(v));
}

__global__ void edge_score_kernel(const int* __restrict__ src, const int* __restrict__ dst,
                                  const float* __restrict__ w_src, const float* __restrict__ w_dst,
                                  float* __restrict__ elist, float* seg_max, int E) {
    const int wave = (blockIdx.x * blockDim.x + threadIdx.x) >> 5;
    if (wave >= E) return;
    const int lane = threadIdx.x & 31;
    const int s = src[wave];
    const int d = dst[wave];
    const float* a = w_src + (size_t)s * OUT_FEATS;
    const float* b = w_dst + (size_t)d * OUT_FEATS;
    float sum = 0.f;
#pragma unroll
    for (int i = 0; i < OUT_FEATS / 32; ++i)
        sum += a[lane + 32 * i] * b[lane + 32 * i];
#pragma unroll
    for (int off = 16; off > 0; off >>= 1)
        sum += __shfl_xor(sum, off, 32);
    if (lane == 0) {
        float sc = sum > 0.f ? sum : NEG_SLOPE * sum;
        elist[wave] = sc;
        atomicMaxFloat(seg_max + d, sc);
    }
}

__global__ void edge_exp_kernel(const int* __restrict__ dst, float* __restrict__ elist,
                                const float* __restrict__ seg_max, float* seg_sum, int E) {
    int e = blockIdx.x * blockDim.x + threadIdx.x;
    if (e >= E) return;
    const int d = dst[e];
    float m = seg_max[d];
    if (!isfinite(m)) m = 0.f;
    const float ex = expf(elist[e] - m);
    elist[e] = ex;
    atomicAdd(seg_sum + d, ex);
}

__global__ void edge_div_kernel(const int* __restrict__ dst, float* __restrict__ elist,
                                const float* __restrict__ seg_sum, int E) {
    int e = blockIdx.x * blockDim.x + threadIdx.x;
    if (e >= E) return;
    elist[e] = elist[e] / seg_sum[dst[e]];
}

// ---------------------------------------------------------------------------
// message passing: h[dst] += feat[src] * norm_l[src]  (one block per edge, 128 feats)
// ---------------------------------------------------------------------------
__global__ void scatter_h_kernel(const int* __restrict__ src, const int* __restrict__ dst,
                                 const float* __restrict__ feat, const float* __restrict__ norm_l,
                                 float* h, int E) {
    const int e = blockIdx.x;
    const int f = threadIdx.x;
    const int u = src[e];
    const int v = dst[e];
    atomicAdd(&h[(size_t)v * IN_FEATS + f], feat[(size_t)u * IN_FEATS + f] * norm_l[u]);
}

// ---------------------------------------------------------------------------
extern "C" void kernel_launch(void* const* d_in, const int* in_sizes, int n_in,
                              void* d_out, int out_size, void* d_ws, size_t ws_size,
                              hipStream_t stream) {
    (void)n_in; (void)out_size; (void)ws_size;
    const float* feat = (const float*)d_in[0];     // [N, 128]
    const float* W    = (const float*)d_in[1];     // [128, 256]
    const int*   src  = (const int*)d_in[2];       // [E]
    const int*   dst  = (const int*)d_in[3];       // [E]
    const int N = in_sizes[0] / IN_FEATS;          // 50000
    const int E = in_sizes[2];                     // 800000

    float* rst   = (float*)d_out;                  // [N, 256]
    float* elist = rst + (size_t)N * OUT_FEATS;    // [E] (scores in-place -> e_soft)

    float* ws      = (float*)d_ws;
    float* norm_l  = ws;                           // [N] out_deg -> rsqrt
    float* norm_r  = norm_l + N;                   // [N] in_deg  -> rsqrt
    float* seg_max = norm_r + N;                   // [N]
    float* seg_sum = seg_max + N;                  // [N]
    float* Wt      = seg_sum + N;                  // [256,128] transposed weight
    float* w_src   = Wt + (size_t)IN_FEATS * OUT_FEATS;  // [N, 256]
    float* w_dst   = w_src + (size_t)N * OUT_FEATS;      // [N, 256]
    float* h       = w_dst + (size_t)N * OUT_FEATS;      // [N, 128]

    const int hsize = N * IN_FEATS;
    init_ws_kernel<<<(hsize + 255) / 256, 256, 0, stream>>>(norm_l, norm_r, seg_max, seg_sum, h, N, hsize);
    transpose_w_kernel<<<(IN_FEATS * OUT_FEATS + 255) / 256, 256, 0, stream>>>(W, Wt);
    degree_kernel<<<(E + 255) / 256, 256, 0, stream>>>(src, dst, norm_l, norm_r, E);
    norm_kernel<<<(N + 255) / 256, 256, 0, stream>>>(norm_l, norm_r, N);

    // 50000/80 = 625 row blocks, 256/32 = 8 col blocks -> 5000 waves
    const int waves = (N / (16 * MB)) * (OUT_FEATS / (16 * NB));
    gemm_proj_kernel<<<(waves + 7) / 8, 256, 0, stream>>>(feat, Wt, norm_l, w_src, w_dst, N);

    edge_score_kernel<<<(E + 7) / 8, 256, 0, stream>>>(src, dst, w_src, w_dst, elist, seg_max, E);
    edge_exp_kernel<<<(E + 255) / 256, 256, 0, stream>>>(dst, elist, seg_max, seg_sum, E);
    edge_div_kernel<<<(E + 255) / 256, 256, 0, stream>>>(dst, elist, seg_sum, E);

    scatter_h_kernel<<<E, IN_FEATS, 0, stream>>>(src, dst, feat, norm_l, h, E);

    gemm_out_kernel<<<(waves + 7) / 8, 256, 0, stream>>>(h, Wt, norm_r, rst, N);
}